// RuleGraphConvLayer_49864570307076
// MI455X (gfx1250) — compile-verified
//
#include <hip/hip_runtime.h>

typedef __attribute__((ext_vector_type(2))) float v2f;
typedef __attribute__((ext_vector_type(8))) float v8f;

#define N_NODES_C   262144
#define N_COLS_C    103
#define NUM_F_C     81
#define OUT_CH_C    128
#define KPAD_C      84      // self region padded 81 -> 84 (multiple of 4)
#define KTOT_C      168     // [self pad84 | comb pad84]
#define A_STRIDE_C  170     // LDS row stride (dwords): even for b64 align, 2-way max conflict
#define WP_STRIDE_C 256     // packed-W K-pair stride in dwords (128 cols * 2)
#define NODES_PB_C  128
#define THREADS_C   256

__global__ __launch_bounds__(THREADS_C)
void rulegraphconv_wmma_f32(const float* __restrict__ feat,
                            const float* __restrict__ w_s,
                            const float* __restrict__ w_n,
                            float* __restrict__ out)
{
    extern __shared__ float smem[];
    // Packed weights: sWp[(k>>1)*256 + 2*col + (k&1)]  (KTOT_C/2 pairs)
    float* sWp  = smem;                                   // (KTOT_C/2) * WP_STRIDE_C
    float* sA   = sWp + (KTOT_C / 2) * WP_STRIDE_C;       // NODES_PB_C * A_STRIDE_C
    int*   sSel = (int*)(sA + NODES_PB_C * A_STRIDE_C);   // NODES_PB_C
    float* sInv = (float*)(sSel + NODES_PB_C);            // NODES_PB_C
    int*   sVal = (int*)(sInv + NODES_PB_C);              // NODES_PB_C

    const int tid       = threadIdx.x;
    const int blockBase = blockIdx.x * NODES_PB_C;

    // ---- Stage combined padded weights [w_s ; w_n[:81]] into LDS, K-pair interleaved ----
    for (int i = tid; i < KTOT_C * OUT_CH_C; i += THREADS_C) {
        int r = i >> 7;              // K row 0..167
        int c = i & (OUT_CH_C - 1);  // output channel
        float v = 0.f;
        if (r < KPAD_C) {
            if (r < NUM_F_C) v = w_s[r * OUT_CH_C + c];
        } else {
            int rr = r - KPAD_C;
            if (rr < NUM_F_C) v = w_n[rr * OUT_CH_C + c];
        }
        sWp[(r >> 1) * WP_STRIDE_C + (c << 1) + (r & 1)] = v;
    }

    // ---- Per-node neighbor selection and 1/d^2 ----
    if (tid < NODES_PB_C) {
        int node = blockBase + tid;
        const float* fr = feat + (size_t)node * N_COLS_C;
        int i0 = (int)fr[NUM_F_C];
        int i1 = (int)fr[NUM_F_C + 1];
        int sel   = (i1 != 0) ? i1 : i0;
        int valid = (i1 != 0) || (i0 != 0);
        float inv = 0.f;
        if (valid) {
            const float* nr = feat + (size_t)sel * N_COLS_C;
            float dx = fr[0] - nr[0];
            float dy = fr[1] - nr[1];
            float dz = fr[2] - nr[2];
            float d2 = dx * dx + dy * dy + dz * dz;
            inv = (d2 > 0.f) ? (1.f / d2) : 10000.f;   // d_eff=0.01 -> 1/d_eff^2
        }
        sSel[tid] = sel;
        sInv[tid] = inv;
        sVal[tid] = valid;
    }
    __syncthreads();

    // ---- Build per-node K-vectors in LDS (coalesced over k) ----
    for (int i = tid; i < NODES_PB_C * KPAD_C; i += THREADS_C) {
        int m = i / KPAD_C;
        int k = i - m * KPAD_C;
        int node = blockBase + m;
        float sv = (k < NUM_F_C) ? feat[(size_t)node * N_COLS_C + k] : 0.f;
        float cv = 0.f;
        if (k < NUM_F_C) {
            float nv  = feat[(size_t)sSel[m] * N_COLS_C + k];
            float inv = sInv[m];
            cv = (k < 3) ? nv * inv : (sv + nv) * inv;
        }
        float* row = sA + m * A_STRIDE_C;
        row[k]          = sv;
        row[KPAD_C + k] = cv;
    }
    __syncthreads();

    // ---- WMMA phase: each wave -> 16 nodes x 128 channels ----
    const int wave = tid >> 5;
    const int lane = tid & 31;
    const int half = lane >> 4;   // lane half: selects K sub-pair / output row-half
    const int mn   = lane & 15;   // A: row M ; B/D: col N

    const float* aRow = sA + (wave * 16 + mn) * A_STRIDE_C;

    v8f acc[8];
    v8f zero = {0.f, 0.f, 0.f, 0.f, 0.f, 0.f, 0.f, 0.f};
    #pragma unroll
    for (int nt = 0; nt < 8; ++nt) acc[nt] = zero;

    // Lane's K elements for BOTH A and B fragments: K = 4t + 2*half + v (v=0,1)
    #pragma unroll 7
    for (int t = 0; t < KTOT_C / 4; ++t) {
        int kk = 4 * t + 2 * half;
        v2f a = *(const v2f*)(aRow + kk);
        const float* wp = sWp + (2 * t + half) * WP_STRIDE_C + (mn << 1);
        #pragma unroll
        for (int nt = 0; nt < 8; ++nt) {
            v2f b = *(const v2f*)(wp + nt * 32);   // pair (K=kk, K=kk+1) adjacent -> b64
            acc[nt] = __builtin_amdgcn_wmma_f32_16x16x4_f32(
                /*neg_a=*/false, a, /*neg_b=*/false, b,
                /*c_mod=*/(short)0, acc[nt],
                /*reuse_a=*/false, /*reuse_b=*/false);
        }
    }

    // ---- Store with validity mask (invalid rows must be exactly zero) ----
    #pragma unroll
    for (int g = 0; g < 8; ++g) {
        int m     = g + 8 * half;          // D layout: VGPR g -> row g (+8 for upper half)
        int local = wave * 16 + m;
        float keep = sVal[local] ? 1.f : 0.f;
        float* orow = out + (size_t)(blockBase + local) * OUT_CH_C + mn;
        #pragma unroll
        for (int nt = 0; nt < 8; ++nt) {
            orow[nt * 16] = acc[nt][g] * keep;
        }
    }
}

extern "C" void kernel_launch(void* const* d_in, const int* in_sizes, int n_in,
                              void* d_out, int out_size, void* d_ws, size_t ws_size,
                              hipStream_t stream) {
    const float* feat = (const float*)d_in[0];
    const float* w_s  = (const float*)d_in[1];
    const float* w_n  = (const float*)d_in[2];
    float* out = (float*)d_out;

    size_t shmem = (size_t)((KTOT_C / 2) * WP_STRIDE_C + NODES_PB_C * A_STRIDE_C) * sizeof(float)
                 + (size_t)NODES_PB_C * (2 * sizeof(int) + sizeof(float));

    dim3 grid(N_NODES_C / NODES_PB_C);   // 2048 blocks
    dim3 block(THREADS_C);
    rulegraphconv_wmma_f32<<<grid, block, shmem, stream>>>(feat, w_s, w_n, out);
}